// NeighbourLoss_69054484185806
// MI455X (gfx1250) — compile-verified
//
#include <hip/hip_runtime.h>

// CDNA5 / gfx1250: wave32, WMMA (not MFMA).
typedef __attribute__((ext_vector_type(2))) float v2f;
typedef __attribute__((ext_vector_type(8))) float v8f;
typedef __attribute__((ext_vector_type(4))) float v4f;
typedef __attribute__((ext_vector_type(4))) int   v4i;

#define NPTS    1048576
#define KNBR    16
#define NTOTAL  (NPTS * KNBR)        // 16,777,216 pair terms
#define NCHUNKS (NTOTAL / 4)         // 4,194,304 chunks of 4 elements
#define TPB     256
#define NBLK    2048
#define NTHREADS (NBLK * TPB)        // 524,288
#define CPT     (NCHUNKS / NTHREADS) // exactly 8 chunks per thread

// Full wave32 f32 sum via two chained V_WMMA_F32_16X16X4_F32 ops.
//
// WMMA #1: A (16x4) with per-lane partial in VGPR0, zero in VGPR1; B = ones.
//   Each (VGPR, lane-half) slot of A maps to a distinct K row, so
//   D[m][n] = p_m + p_{m+16}. Lane l's 8 D VGPRs then hold the row-sums r_M
//   for one half of the 16 M rows.
// WMMA #2: fold d[0..3] -> b0 and d[4..7] -> b1 and use {b0,b1} as the
//   B matrix (4x16) with A = ones. Each column sum covers all four
//   (VGPR, lane-half) slots exactly once = the total wave sum, broadcast to
//   every lane/VGPR of D2. Robust to any bijective slot->K row permutation.
//
// Exact f32 RNE arithmetic; requires EXEC all-ones (uniform call sites only).
__device__ __forceinline__ float wave_sum_wmma(float p) {
  const v2f ones = {1.0f, 1.0f};
  v2f a; a[0] = p; a[1] = 0.0f;
  v8f c = {};
  v8f d = __builtin_amdgcn_wmma_f32_16x16x4_f32(
      false, a, false, ones, (short)0, c, false, false);
  v2f b; b[0] = (d[0] + d[1]) + (d[2] + d[3]);
         b[1] = (d[4] + d[5]) + (d[6] + d[7]);
  v8f d2 = __builtin_amdgcn_wmma_f32_16x16x4_f32(
      false, ones, false, b, (short)0, c, false, false);
  return d2[0];   // full wave sum in every lane
}

__global__ void __launch_bounds__(TPB)
nloss_partial_kernel(const float* __restrict__ points,
                     const int*   __restrict__ nbr,
                     const float* __restrict__ odist,
                     float*       __restrict__ partials) {
  const v4i* __restrict__ nbr4 = (const v4i*)nbr;
  const v4f* __restrict__ od4  = (const v4f*)odist;

  const int tid = blockIdx.x * TPB + threadIdx.x;
  float acc = 0.0f;

#pragma unroll
  for (int i = 0; i < CPT; ++i) {
    const int chunk = tid + i * NTHREADS;    // exact cover, no bounds check
    // 4 chunks per center point (K = 16): all 4 elements share one center.
    const int n = chunk >> 2;
    const float* __restrict__ cp = points + 3 * n;   // broadcast-shared, RT/L2
    const float cx = cp[0], cy = cp[1], cz = cp[2];

    // 64 MB streams: 16B non-temporal loads -> don't evict L2-resident points.
    const v4i idx = __builtin_nontemporal_load(nbr4 + chunk);
    const v4f od  = __builtin_nontemporal_load(od4  + chunk);

#pragma unroll
    for (int j = 0; j < 4; ++j) {
      const float* __restrict__ q = points + 3 * idx[j];  // L2-resident gather
      const float dx = cx - q[0];
      const float dy = cy - q[1];
      const float dz = cz - q[2];
      const float cur = fmaf(dx, dx, fmaf(dy, dy, dz * dz));
      const float err = cur - od[j];
      acc = fmaf(err, err, acc);
    }
  }

  // Block reduction: WMMA wave sums -> 8 wave partials in LDS -> thread 0.
  __shared__ float wpart[TPB / 32];
  const float wsum = wave_sum_wmma(acc);     // uniform call, EXEC all-1s
  if ((threadIdx.x & 31) == 0) wpart[threadIdx.x >> 5] = wsum;
  __syncthreads();
  if (threadIdx.x == 0) {
    float b = 0.0f;
#pragma unroll
    for (int i = 0; i < TPB / 32; ++i) b += wpart[i];
    partials[blockIdx.x] = b;
  }
}

__global__ void __launch_bounds__(TPB)
nloss_final_kernel(const float* __restrict__ partials,
                   float*       __restrict__ out) {
  float acc = 0.0f;
#pragma unroll
  for (int i = 0; i < NBLK / TPB; ++i)       // exactly 8, no bounds check
    acc += partials[threadIdx.x + i * TPB];

  __shared__ float wpart[TPB / 32];
  const float wsum = wave_sum_wmma(acc);
  if ((threadIdx.x & 31) == 0) wpart[threadIdx.x >> 5] = wsum;
  __syncthreads();
  if (threadIdx.x == 0) {
    float b = 0.0f;
#pragma unroll
    for (int i = 0; i < TPB / 32; ++i) b += wpart[i];
    // mean * LOSS_SCALE = sum * (1e5 / (N*K))
    out[0] = b * (float)(100000.0 / (double)NTOTAL);
  }
}

extern "C" void kernel_launch(void* const* d_in, const int* in_sizes, int n_in,
                              void* d_out, int out_size, void* d_ws, size_t ws_size,
                              hipStream_t stream) {
  const float* points = (const float*)d_in[0];   // [N,3] f32
  const int*   nbr    = (const int*)d_in[1];     // [N,K] i32
  const float* odist  = (const float*)d_in[2];   // [N*K] f32
  float* out      = (float*)d_out;               // scalar f32
  float* partials = (float*)d_ws;                // 2048 f32 = 8 KB scratch

  nloss_partial_kernel<<<NBLK, TPB, 0, stream>>>(points, nbr, odist, partials);
  nloss_final_kernel<<<1, TPB, 0, stream>>>(partials, out);
}